// BaseHMM_75196287418428
// MI455X (gfx1250) — compile-verified
//
#include <hip/hip_runtime.h>
#include <math.h>

// HMM forward filter, exact 3-phase parallel scan for MI455X (gfx1250).
//   T = 1e6 steps, K = 64 states, chunked as C=4000 chunks x L=250 steps.
// Outputs (f32, concatenated): ut[T*64], u_norm[T*64], ft[T]  (516 MB).
// Workspace layout (floats): P[4096] | pi[64] | vstart[4000*64] | Mbuf[4000*4096]
//   total ~66.6 MB.

#define HMM_T 1000000
#define HMM_K 64
#define HMM_C 4000
#define HMM_L 250
#define LSTR 65  // padded LDS row stride (floats) -> bank = (row+col)%64
#define INV_SQRT_2PI 0.3989422804014327f

typedef __attribute__((ext_vector_type(2))) float v2f;
typedef __attribute__((ext_vector_type(8))) float v8f;

static __device__ __forceinline__ v8f wmma_f32_16x16x4(v2f a, v2f b, v8f c) {
  // D(16x16,f32) = A(16x4,f32) x B(4x16,f32) + C
  return __builtin_amdgcn_wmma_f32_16x16x4_f32(false, a, false, b, (short)0, c,
                                               false, false);
}

// ---------------------------------------------------------------------------
// Kernel 0: row softmax of transition logits -> P (row-stochastic 64x64)
// ---------------------------------------------------------------------------
__global__ __launch_bounds__(64) void hmm_softmax_rows(
    const float* __restrict__ logits, float* __restrict__ P) {
  __shared__ float red[64];
  const int row = blockIdx.x, tid = threadIdx.x;
  float v = logits[row * 64 + tid];
  red[tid] = v;
  __syncthreads();
  for (int off = 32; off >= 1; off >>= 1) {
    if (tid < off) red[tid] = fmaxf(red[tid], red[tid + off]);
    __syncthreads();
  }
  float mx = red[0];
  __syncthreads();
  float e = __expf(v - mx);
  red[tid] = e;
  __syncthreads();
  for (int off = 32; off >= 1; off >>= 1) {
    if (tid < off) red[tid] += red[tid + off];
    __syncthreads();
  }
  P[row * 64 + tid] = e / red[0];
}

// ---------------------------------------------------------------------------
// Kernel 1: stationary distribution pi: solve (P^T - I, last row = 1) x = e63
// Gaussian elimination w/ partial pivoting in LDS, one 64-thread block.
// ---------------------------------------------------------------------------
__global__ __launch_bounds__(64) void hmm_stationary(const float* __restrict__ P,
                                                     float* __restrict__ pi) {
  __shared__ float A[64 * 65];  // augmented [A|b]
  __shared__ float xs[64];
  __shared__ int piv;
  const int tid = threadIdx.x;
  for (int r = 0; r < 64; ++r)
    A[r * 65 + tid] = P[tid * 64 + r] - (r == tid ? 1.0f : 0.0f);
  A[63 * 65 + tid] = 1.0f;
  if (tid == 0)
    for (int r = 0; r < 64; ++r) A[r * 65 + 64] = (r == 63) ? 1.0f : 0.0f;
  __syncthreads();
  for (int k = 0; k < 64; ++k) {
    if (tid == 0) {
      int p = k;
      float best = fabsf(A[k * 65 + k]);
      for (int r = k + 1; r < 64; ++r) {
        float m = fabsf(A[r * 65 + k]);
        if (m > best) { best = m; p = r; }
      }
      piv = p;
    }
    __syncthreads();
    const int p = piv;
    if (p != k) {
      float t0 = A[k * 65 + tid];
      A[k * 65 + tid] = A[p * 65 + tid];
      A[p * 65 + tid] = t0;
      if (tid == 0) {
        float t1 = A[k * 65 + 64];
        A[k * 65 + 64] = A[p * 65 + 64];
        A[p * 65 + 64] = t1;
      }
    }
    __syncthreads();
    if (tid > k) {
      float f = A[tid * 65 + k] / A[k * 65 + k];
      for (int c = k; c <= 64; ++c) A[tid * 65 + c] -= f * A[k * 65 + c];
    }
    __syncthreads();
  }
  if (tid == 0) {
    for (int k = 63; k >= 0; --k) {
      float s = A[k * 65 + 64];
      for (int c = k + 1; c < 64; ++c) s -= A[k * 65 + c] * xs[c];
      xs[k] = s / A[k * 65 + k];
    }
  }
  __syncthreads();
  pi[tid] = xs[tid];
}

// ---------------------------------------------------------------------------
// Kernel 2 (phase 1): per-chunk transfer matrix M_c = prod_t P*diag(g_t),
// renormalized each step. One wave per chunk; f32 WMMA 16x16x4; P kept in
// B-layout registers, running product in A-layout registers, D->A relayout
// through padded LDS. The 4 column tiles of each row strip are accumulated
// with interleaved accumulators (dep distance 4) to hide WMMA latency.
//  A-layout (16x4): lane = row (both halves), elem vv -> K = 4*kk + 2*half + vv
//  B-layout (4x16): lane&15 = col,            elem vv -> K = 4*kk + 2*half + vv
//  D-layout (16x16): elem v -> row = v + 8*half, lane&15 = col
// ---------------------------------------------------------------------------
__global__ __launch_bounds__(32) void hmm_chunk_products(
    const float* __restrict__ P, const float* __restrict__ y,
    const float* __restrict__ mu, const float* __restrict__ log_sigma,
    float* __restrict__ Mbuf) {
  __shared__ float lds[64 * LSTR];
  const int lane = threadIdx.x;
  const int n = lane & 15, h = lane >> 4;
  const int c = blockIdx.x;

  // per-lane emission params for its 4 columns (16*j + n)
  float mu4[4], is4[4], in4[4];
#pragma unroll
  for (int j = 0; j < 4; ++j) {
    int col = 16 * j + n;
    mu4[j] = mu[col];
    float e = __expf(-log_sigma[col]);  // 1/sigma
    is4[j] = e;
    in4[j] = e * INV_SQRT_2PI;
  }
  // P in B layout (128 regs/lane)
  float pb[16][4][2];
#pragma unroll
  for (int kk = 0; kk < 16; ++kk)
#pragma unroll
    for (int j = 0; j < 4; ++j)
#pragma unroll
      for (int vv = 0; vv < 2; ++vv)
        pb[kk][j][vv] = P[(4 * kk + 2 * h + vv) * 64 + 16 * j + n];
  // running product M = I in A layout (128 regs/lane)
  float ma[4][16][2];
#pragma unroll
  for (int it = 0; it < 4; ++it)
#pragma unroll
    for (int kk = 0; kk < 16; ++kk)
#pragma unroll
      for (int vv = 0; vv < 2; ++vv)
        ma[it][kk][vv] =
            ((16 * it + n) == (4 * kk + 2 * h + vv)) ? 1.0f : 0.0f;

  const size_t t0 = (size_t)c * HMM_L;
  float inv = 1.0f;
  for (int i = 0; i < HMM_L; ++i) {
    const float yt = y[t0 + i];
    float g4[4];
#pragma unroll
    for (int j = 0; j < 4; ++j) {
      float z = (yt - mu4[j]) * is4[j];
      g4[j] = __expf(-0.5f * z * z) * in4[j];
    }
    float s = 0.0f;
#pragma unroll
    for (int it = 0; it < 4; ++it) {
      // 4 interleaved accumulator chains across the column tiles
      v8f acc[4] = {{}, {}, {}, {}};
#pragma unroll
      for (int kk = 0; kk < 16; ++kk) {
        v2f a = {ma[it][kk][0], ma[it][kk][1]};
#pragma unroll
        for (int j = 0; j < 4; ++j) {
          v2f b = {pb[kk][j][0], pb[kk][j][1]};
          acc[j] = wmma_f32_16x16x4(a, b, acc[j]);
        }
      }
#pragma unroll
      for (int j = 0; j < 4; ++j)
#pragma unroll
        for (int v = 0; v < 8; ++v) {
          float w = acc[j][v] * g4[j];  // column scale by emission
          s += w;
          lds[(16 * it + v + 8 * h) * LSTR + 16 * j + n] = w;
        }
    }
    // wave32 total sum -> per-step renormalization (keeps chain in f32 range)
    s += __shfl_xor(s, 1, 32);
    s += __shfl_xor(s, 2, 32);
    s += __shfl_xor(s, 4, 32);
    s += __shfl_xor(s, 8, 32);
    s += __shfl_xor(s, 16, 32);
    inv = 1.0f / s;
    __syncthreads();
    // reload M in A layout, normalized
#pragma unroll
    for (int it = 0; it < 4; ++it)
#pragma unroll
      for (int kk = 0; kk < 16; ++kk)
#pragma unroll
        for (int vv = 0; vv < 2; ++vv)
          ma[it][kk][vv] =
              lds[(16 * it + n) * LSTR + 4 * kk + 2 * h + vv] * inv;
    __syncthreads();
  }
  // write normalized M_c row-major (coalesced from LDS); keep RT caching so
  // the 64 MB Mbuf stays resident in the 192 MB L2 for phase 2.
  float* out = Mbuf + (size_t)c * 4096;
  for (int r = 0; r < 64; ++r) {
    out[r * 64 + lane] = lds[r * LSTR + lane] * inv;
    out[r * 64 + lane + 32] = lds[r * LSTR + lane + 32] * inv;
  }
}

// ---------------------------------------------------------------------------
// Kernel 3 (phase 2): sequential chunk-level scan of start vectors.
//   vstart[0] = pi (since pi @ P = pi, making t=0 uniform with the recurrence)
//   vstart[c+1] = normalize(vstart[c] @ M_c)
// One 256-thread block; L2-resident M_c double-buffered through LDS.
// ---------------------------------------------------------------------------
__global__ __launch_bounds__(256) void hmm_chunk_scan(
    const float* __restrict__ pi, const float* __restrict__ Mbuf,
    float* __restrict__ vstart) {
  __shared__ float Mb[2][4096];
  __shared__ float vb[64];
  __shared__ float part[256];
  __shared__ float colv[64];
  __shared__ float sinv;
  const int tid = threadIdx.x;
  const int j = tid & 63, q = tid >> 6;
  if (tid < 64) {
    float p = pi[tid];
    vb[tid] = p;
    vstart[tid] = p;
  }
#pragma unroll
  for (int s = 0; s < 16; ++s) Mb[0][tid + 256 * s] = Mbuf[tid + 256 * s];
  __syncthreads();
  for (int c = 0; c < HMM_C - 1; ++c) {
    const int cur = c & 1;
    float nxt[16];
    const size_t base = (size_t)(c + 1) * 4096;
#pragma unroll
    for (int s = 0; s < 16; ++s) nxt[s] = Mbuf[base + tid + 256 * s];
    float p = 0.0f;
#pragma unroll
    for (int kk = 0; kk < 16; ++kk) {
      int k = q * 16 + kk;
      p += vb[k] * Mb[cur][k * 64 + j];
    }
    part[tid] = p;
    __syncthreads();
    if (tid < 64)
      colv[j] = part[j] + part[j + 64] + part[j + 128] + part[j + 192];
    __syncthreads();
    if (tid < 32) {  // wave-level total reduction (no extra barriers)
      float s = colv[tid] + colv[tid + 32];
      s += __shfl_xor(s, 1, 32);
      s += __shfl_xor(s, 2, 32);
      s += __shfl_xor(s, 4, 32);
      s += __shfl_xor(s, 8, 32);
      s += __shfl_xor(s, 16, 32);
      if (tid == 0) sinv = 1.0f / s;
    }
    __syncthreads();
    if (tid < 64) {
      float nv = colv[j] * sinv;
      vb[j] = nv;
      vstart[(size_t)(c + 1) * 64 + j] = nv;
    }
#pragma unroll
    for (int s = 0; s < 16; ++s) Mb[cur ^ 1][tid + 256 * s] = nxt[s];
    __syncthreads();
  }
}

// ---------------------------------------------------------------------------
// Kernel 4 (phase 3): exact per-step replay + output streaming.
// One wave handles 16 consecutive chunks as the 16 rows of the WMMA A operand:
//   U_new = normalize_rows((U @ P) .* g[t_row, :]),  64 WMMAs/step with the
// 4 column-tile accumulators interleaved (dep distance 4). Streams ut /
// u_norm / ft (516 MB total) with non-temporal 16-lane-contiguous stores so
// write-once output lines don't evict read-shared data from L2.
// ---------------------------------------------------------------------------
__global__ __launch_bounds__(32) void hmm_filter_chunks(
    const float* __restrict__ P, const float* __restrict__ y,
    const float* __restrict__ mu, const float* __restrict__ log_sigma,
    const float* __restrict__ vstart, float* __restrict__ out_ut,
    float* __restrict__ out_un, float* __restrict__ out_ft) {
  __shared__ float uls[16 * LSTR];
  __shared__ float ybuf[16];
  const int lane = threadIdx.x;
  const int n = lane & 15, h = lane >> 4;
  const int c0 = blockIdx.x * 16;

  float mu4[4], is4[4], in4[4];
#pragma unroll
  for (int j = 0; j < 4; ++j) {
    int col = 16 * j + n;
    mu4[j] = mu[col];
    float e = __expf(-log_sigma[col]);
    is4[j] = e;
    in4[j] = e * INV_SQRT_2PI;
  }
  float pb[16][4][2];
#pragma unroll
  for (int kk = 0; kk < 16; ++kk)
#pragma unroll
    for (int j = 0; j < 4; ++j)
#pragma unroll
      for (int vv = 0; vv < 2; ++vv)
        pb[kk][j][vv] = P[(4 * kk + 2 * h + vv) * 64 + 16 * j + n];
  float ua[16][2];  // U (16 rows x 64) in A layout; row n <-> chunk c0+n
#pragma unroll
  for (int kk = 0; kk < 16; ++kk)
#pragma unroll
    for (int vv = 0; vv < 2; ++vv)
      ua[kk][vv] = vstart[(size_t)(c0 + n) * 64 + 4 * kk + 2 * h + vv];

  for (int i = 0; i < HMM_L; ++i) {
    if (lane < 16) ybuf[lane] = y[(size_t)(c0 + lane) * HMM_L + i];
    __syncthreads();
    // D = U @ P with 4 interleaved accumulator chains
    v8f d[4] = {{}, {}, {}, {}};
#pragma unroll
    for (int kk = 0; kk < 16; ++kk) {
      v2f a = {ua[kk][0], ua[kk][1]};
#pragma unroll
      for (int j = 0; j < 4; ++j) {
        v2f b = {pb[kk][j][0], pb[kk][j][1]};
        d[j] = wmma_f32_16x16x4(a, b, d[j]);
      }
    }
    float wv[4][8];
    float pr[8];
#pragma unroll
    for (int v = 0; v < 8; ++v) pr[v] = 0.0f;
#pragma unroll
    for (int j = 0; j < 4; ++j)
#pragma unroll
      for (int v = 0; v < 8; ++v) {
        const int r = v + 8 * h;
        const size_t t = (size_t)(c0 + r) * HMM_L + i;
        float utv = d[j][v];
        __builtin_nontemporal_store(utv, &out_ut[t * 64 + 16 * j + n]);
        float z = (ybuf[r] - mu4[j]) * is4[j];
        float g = __expf(-0.5f * z * z) * in4[j];
        float w = utv * g;
        wv[j][v] = w;
        pr[v] += w;
      }
    // row sums (ft) via xor-shuffles within each 16-lane half
#pragma unroll
    for (int v = 0; v < 8; ++v) {
      float s = pr[v];
      s += __shfl_xor(s, 1, 32);
      s += __shfl_xor(s, 2, 32);
      s += __shfl_xor(s, 4, 32);
      s += __shfl_xor(s, 8, 32);
      pr[v] = s;
    }
    if (n == 0) {
#pragma unroll
      for (int v = 0; v < 8; ++v)
        __builtin_nontemporal_store(
            pr[v], &out_ft[(size_t)(c0 + v + 8 * h) * HMM_L + i]);
    }
    float iv[8];
#pragma unroll
    for (int v = 0; v < 8; ++v) iv[v] = 1.0f / pr[v];
#pragma unroll
    for (int j = 0; j < 4; ++j)
#pragma unroll
      for (int v = 0; v < 8; ++v) {
        const int r = v + 8 * h;
        const size_t t = (size_t)(c0 + r) * HMM_L + i;
        float un = wv[j][v] * iv[v];
        __builtin_nontemporal_store(un, &out_un[t * 64 + 16 * j + n]);
        uls[r * LSTR + 16 * j + n] = un;  // for D->A relayout
      }
    __syncthreads();
#pragma unroll
    for (int kk = 0; kk < 16; ++kk)
#pragma unroll
      for (int vv = 0; vv < 2; ++vv)
        ua[kk][vv] = uls[n * LSTR + 4 * kk + 2 * h + vv];
    __syncthreads();
  }
}

// ---------------------------------------------------------------------------
extern "C" void kernel_launch(void* const* d_in, const int* in_sizes, int n_in,
                              void* d_out, int out_size, void* d_ws,
                              size_t ws_size, hipStream_t stream) {
  (void)in_sizes; (void)n_in; (void)out_size; (void)ws_size;
  const float* y = (const float*)d_in[0];
  const float* logits = (const float*)d_in[1];
  const float* mu = (const float*)d_in[2];
  const float* lsg = (const float*)d_in[3];

  float* ws = (float*)d_ws;
  float* P = ws;                                   // 4096
  float* pi = ws + 4096;                           // 64
  float* vstart = ws + 4096 + 64;                  // HMM_C*64
  float* Mbuf = vstart + (size_t)HMM_C * 64;       // HMM_C*4096 (~64 MB)

  float* out_ut = (float*)d_out;
  float* out_un = out_ut + (size_t)HMM_T * 64;
  float* out_ft = out_ut + (size_t)2 * HMM_T * 64;

  hmm_softmax_rows<<<64, 64, 0, stream>>>(logits, P);
  hmm_stationary<<<1, 64, 0, stream>>>(P, pi);
  hmm_chunk_products<<<HMM_C, 32, 0, stream>>>(P, y, mu, lsg, Mbuf);
  hmm_chunk_scan<<<1, 256, 0, stream>>>(pi, Mbuf, vstart);
  hmm_filter_chunks<<<HMM_C / 16, 32, 0, stream>>>(P, y, mu, lsg, vstart,
                                                   out_ut, out_un, out_ft);
}